// KNNGetterLayer_43104291783208
// MI455X (gfx1250) — compile-verified
//
#include <hip/hip_runtime.h>

typedef __attribute__((ext_vector_type(2))) float v2f;
typedef __attribute__((ext_vector_type(8))) float v8f;

#define N_IN    32768
#define N_OUT   8192
#define KSEL    32
#define TILES   (N_IN / 16)      // 2048 total 16-sample tiles
#define NWAVES  4                // waves per block, each scans TILES/NWAVES tiles
#define WTILES  (TILES / NWAVES) // 512 tiles per wave
#define WIN     8                // flush window (tiles)
#define CAP     (WIN * 16)       // 128: max possible pushes per window -> never overflows

__global__ __launch_bounds__(32 * NWAVES)
void knn_wmma_kernel(const float* __restrict__ samples,
                     const float* __restrict__ outs,
                     long long* __restrict__ outIdx,
                     float* __restrict__ outDist) {
    __shared__ float    bufK[NWAVES][16][CAP];   // 32 KB
    __shared__ unsigned bufI[NWAVES][16][CAP];   // 32 KB
    __shared__ float    heapK[NWAVES][16][KSEL]; // 8 KB
    __shared__ unsigned heapI[NWAVES][16][KSEL]; // 8 KB
    __shared__ unsigned cnt[NWAVES][16];
    __shared__ float    curMax[NWAVES][16];
    __shared__ int      curMaxPos[NWAVES][16];
    __shared__ float    a2s[16];

    const int  tid    = threadIdx.x;
    const int  lane   = tid & 31;             // 0..31, wave32
    const int  wv     = tid >> 5;             // 0..3
    const bool lo     = lane < 16;
    const int  rowOff = lo ? 0 : 8;           // f32 C/D layout: VGPR r -> row r + 8*(lane>=16)
    const int  nSub   = lane & 15;
    const int  wg     = blockIdx.x;           // 16 query points per block
    const float INF   = __builtin_inff();

    // ---- A tile: 16 queries x K=4 (K=2,3 zero-padded), pre-scaled by -2 ----
    v2f a; a[0] = 0.0f; a[1] = 0.0f;
    if (lo) {
        const float2 q = reinterpret_cast<const float2*>(outs)[wg * 16 + lane];
        a[0] = -2.0f * q.x;
        a[1] = -2.0f * q.y;
        if (wv == 0) a2s[lane] = q.x * q.x + q.y * q.y;
        cnt[wv][lane] = 0u;
        curMax[wv][lane] = INF;
        curMaxPos[wv][lane] = 0;
        #pragma unroll
        for (int j = 0; j < KSEL; ++j) { heapK[wv][lane][j] = INF; heapI[wv][lane][j] = 0u; }
    }
    __syncthreads();

    float thr[8];
    #pragma unroll
    for (int r = 0; r < 8; ++r) thr[r] = INF;
    float tmax = INF;

    const float2* s2   = reinterpret_cast<const float2*>(samples);
    const int     tBeg = wv * WTILES;
    const int     tEnd = tBeg + WTILES;

    // preload first window
    float2 sv[WIN];
    #pragma unroll
    for (int u = 0; u < WIN; ++u)
        sv[u] = s2[((tBeg + u) << 4) + nSub];

    for (int tw = tBeg; tw < tEnd; tw += WIN) {
        // issue NEXT window's loads now: they fly during this window's WMMA work
        // and across the flush barrier (barrier only forces dscnt waits)
        const int twn = tw + WIN;
        float2 nv[WIN];
        if (twn < tEnd) {
            if (twn + WIN < tEnd)
                __builtin_prefetch(&s2[(twn + WIN) << 4], 0, 1);   // global_prefetch_b8
            #pragma unroll
            for (int u = 0; u < WIN; ++u)
                nv[u] = s2[((twn + u) << 4) + nSub];
        } else {
            #pragma unroll
            for (int u = 0; u < WIN; ++u) nv[u] = sv[u];
        }

        #pragma unroll
        for (int u = 0; u < WIN; ++u) {
            const int n = ((tw + u) << 4) + nSub;
            const float2 s = sv[u];
            // B tile 4x16: VGPR0 = K0 row (lanes 0-15) / K2==0; VGPR1 = K1 / K3==0
            v2f b; b[0] = lo ? s.x : 0.0f; b[1] = lo ? s.y : 0.0f;
            const float cv = s.x * s.x + s.y * s.y;           // |b_n|^2
            v8f c = {cv, cv, cv, cv, cv, cv, cv, cv};

            // key[m][n] = |b_n|^2 - 2 a_m . b_n  (== dist^2 - |a_m|^2)
            v8f key = __builtin_amdgcn_wmma_f32_16x16x4_f32(
                false, a, false, b, (short)0, c, false, false);

            // cheap screen: min of the 8 keys vs max of the 8 thresholds
            const float kmin =
                __builtin_fminf(
                    __builtin_fminf(__builtin_fminf(key[0], key[1]),
                                    __builtin_fminf(key[2], key[3])),
                    __builtin_fminf(__builtin_fminf(key[4], key[5]),
                                    __builtin_fminf(key[6], key[7])));
            if (kmin < tmax) {
                #pragma unroll
                for (int r = 0; r < 8; ++r) {
                    const float kv = key[r];
                    if (kv < thr[r]) {
                        const int row = r + rowOff;
                        const unsigned slot = atomicAdd(&cnt[wv][row], 1u); // ds_add_rtn_u32
                        bufK[wv][row][slot] = kv;       // slot < CAP by construction
                        bufI[wv][row][slot] = (unsigned)n;
                    }
                }
            }
        }

        // ---- flush window: 16 lanes drain their row's candidates into the heap ----
        __syncthreads();
        if (lo) {
            const int row = lane;
            unsigned m = cnt[wv][row]; if (m > CAP) m = CAP;
            float mx = curMax[wv][row];
            int   mpos = curMaxPos[wv][row];
            for (unsigned i = 0; i < m; ++i) {
                const float kk = bufK[wv][row][i];
                if (kk < mx) {
                    heapK[wv][row][mpos] = kk;
                    heapI[wv][row][mpos] = bufI[wv][row][i];
                    mx = heapK[wv][row][0]; mpos = 0;
                    #pragma unroll
                    for (int j = 1; j < KSEL; ++j) {
                        const float hj = heapK[wv][row][j];
                        if (hj > mx) { mx = hj; mpos = j; }
                    }
                }
            }
            cnt[wv][row] = 0u;
            curMax[wv][row] = mx;
            curMaxPos[wv][row] = mpos;
        }
        __syncthreads();
        #pragma unroll
        for (int r = 0; r < 8; ++r) thr[r] = curMax[wv][r + rowOff];
        tmax = __builtin_fmaxf(
                   __builtin_fmaxf(__builtin_fmaxf(thr[0], thr[1]),
                                   __builtin_fmaxf(thr[2], thr[3])),
                   __builtin_fmaxf(__builtin_fmaxf(thr[4], thr[5]),
                                   __builtin_fmaxf(thr[6], thr[7])));

        #pragma unroll
        for (int u = 0; u < WIN; ++u) sv[u] = nv[u];
    }

    // ---- merge the 4 per-wave heaps (128 candidates/row) into final top-32 ----
    __syncthreads();
    float*    mK = &bufK[0][0][0];   // reuse candidate buffers: need 16*128 floats
    unsigned* mI = &bufI[0][0][0];
    for (int e = tid; e < NWAVES * 16 * KSEL; e += 32 * NWAVES) {
        const int w   = e >> 9;        // / (16*KSEL)
        const int rem = e & 511;
        const int row = rem >> 5;
        const int q   = rem & 31;
        mK[row * 128 + w * 32 + q] = heapK[w][row][q];
        mI[row * 128 + w * 32 + q] = heapI[w][row][q];
    }
    __syncthreads();

    if (tid < 16) {
        const int row = tid;
        const int col = wg * 16 + row;
        const float a2v = a2s[row];
        for (int j = 0; j < KSEL; ++j) {
            int   best = 0;
            float bk   = mK[row * 128];
            for (int q = 1; q < 128; ++q) {
                const float qk = mK[row * 128 + q];
                if (qk < bk) { bk = qk; best = q; }
            }
            mK[row * 128 + best] = INF;    // remove from pool
            float d2 = bk + a2v;
            d2 = d2 < 0.0f ? 0.0f : d2;
            outIdx[(size_t)j * N_OUT + col]  = (long long)mI[row * 128 + best];
            outDist[(size_t)j * N_OUT + col] = __builtin_sqrtf(d2);
        }
    }
}

extern "C" void kernel_launch(void* const* d_in, const int* in_sizes, int n_in,
                              void* d_out, int out_size, void* d_ws, size_t ws_size,
                              hipStream_t stream) {
    (void)in_sizes; (void)n_in; (void)out_size; (void)d_ws; (void)ws_size;
    const float* samples = (const float*)d_in[0];     // [32768, 2] f32
    const float* outs    = (const float*)d_in[1];     // [8192, 2]  f32
    // d_out = knn_indices [32, 8192] int64, then knn_distances [32, 8192] f32
    long long* outIdx = (long long*)d_out;
    float* outDist = (float*)(outIdx + (size_t)KSEL * N_OUT);
    knn_wmma_kernel<<<dim3(N_OUT / 16), dim3(32 * NWAVES), 0, stream>>>(
        samples, outs, outIdx, outDist);
}